// Discriminator_75067438399560
// MI455X (gfx1250) — compile-verified
//
#include <hip/hip_runtime.h>
#include <hip/hip_bf16.h>

#define NPTS   8192
#define NCTR   1024
#define NBATCH 4
#define NCTOT  (NBATCH * NCTR)   // 4096 centers total

typedef __attribute__((ext_vector_type(2))) float v2f;
typedef __attribute__((ext_vector_type(8))) float v8f;

// ---------------------------------------------------------------------------
// Kernel 1: farthest point sampling. One block per batch, 1024 threads,
// 8 points per thread held in registers. Per step: update min-dist, then
// argmax via wave32 shuffle tree + 32-entry LDS reduction.
// ---------------------------------------------------------------------------
__global__ __launch_bounds__(1024) void fps_kernel(const float* __restrict__ pcd,
                                                   float* __restrict__ nxyz)
{
    __shared__ float swv[32];
    __shared__ int   swi[32];
    __shared__ int   sFar;
    const int b = blockIdx.x;
    const int t = threadIdx.x;
    const float* P = pcd + (size_t)b * NPTS * 3;

    float px[8], py[8], pz[8], dist[8];
#pragma unroll
    for (int j = 0; j < 8; ++j) {
        int i = t + j * 1024;
        px[j] = P[i * 3 + 0];
        py[j] = P[i * 3 + 1];
        pz[j] = P[i * 3 + 2];
        dist[j] = 1e10f;
    }

    int far = 0;
    for (int s = 0; s < NCTR; ++s) {
        const float cx = P[far * 3 + 0];
        const float cy = P[far * 3 + 1];
        const float cz = P[far * 3 + 2];
        if (t == 0) {
            float* d = nxyz + ((size_t)b * NCTR + s) * 3;
            d[0] = cx; d[1] = cy; d[2] = cz;
        }
        float best = -1.0f; int bi = 0;
#pragma unroll
        for (int j = 0; j < 8; ++j) {
            float dx = px[j] - cx, dy = py[j] - cy, dz = pz[j] - cz;
            float d2 = dx * dx + dy * dy + dz * dz;
            dist[j] = fminf(dist[j], d2);
            if (dist[j] > best) { best = dist[j]; bi = t + j * 1024; }
        }
        // wave32 argmax (first-index on ties, matching jnp.argmax)
        float v = best; int vi = bi;
#pragma unroll
        for (int off = 16; off > 0; off >>= 1) {
            float ov = __shfl_xor(v, off, 32);
            int   oi = __shfl_xor(vi, off, 32);
            if (ov > v || (ov == v && oi < vi)) { v = ov; vi = oi; }
        }
        if ((t & 31) == 0) { swv[t >> 5] = v; swi[t >> 5] = vi; }
        __syncthreads();
        if (t < 32) {
            v = swv[t]; vi = swi[t];
#pragma unroll
            for (int off = 16; off > 0; off >>= 1) {
                float ov = __shfl_xor(v, off, 32);
                int   oi = __shfl_xor(vi, off, 32);
                if (ov > v || (ov == v && oi < vi)) { v = ov; vi = oi; }
            }
            if (t == 0) sFar = vi;
        }
        __syncthreads();
        far = sFar;
    }
}

// ---------------------------------------------------------------------------
// Kernel 2: ball query + grouping for all three radii. One wave per center.
// Ordered append (ascending point index) via ballot + prefix popcount.
// Writes centered coordinates; pads tail slots with the first hit.
// ---------------------------------------------------------------------------
__global__ __launch_bounds__(256) void ball_group_kernel(
    const float* __restrict__ pcd, const float* __restrict__ nxyz,
    float* __restrict__ h0, float* __restrict__ h1, float* __restrict__ h2)
{
    const int lane = threadIdx.x & 31;
    const int wv   = threadIdx.x >> 5;
    const int s    = blockIdx.x * 8 + wv;       // 512 blocks x 8 waves = 4096
    const int b    = s >> 10;
    const float* P = pcd + (size_t)b * NPTS * 3;

    const float cx = nxyz[s * 3 + 0];
    const float cy = nxyz[s * 3 + 1];
    const float cz = nxyz[s * 3 + 2];

    const float r0s = 0.1f * 0.1f, r1s = 0.2f * 0.2f, r2s = 0.4f * 0.4f;
    int c0 = 0, c1 = 0, c2 = 0;
    float f0x = 0.f, f0y = 0.f, f0z = 0.f;
    float f1x = 0.f, f1y = 0.f, f1z = 0.f;
    float f2x = 0.f, f2y = 0.f, f2z = 0.f;
    const unsigned ltmask = (1u << lane) - 1u;

    for (int base = 0; base < NPTS; base += 32) {
        if (c0 >= 16 && c1 >= 32 && c2 >= 128) break;   // uniform per wave
        const int i = base + lane;
        const float dx = P[i * 3 + 0] - cx;
        const float dy = P[i * 3 + 1] - cy;
        const float dz = P[i * 3 + 2] - cz;
        const float d2 = dx * dx + dy * dy + dz * dz;

        { // radius 0.4 -> 128 samples
            const bool p = d2 <= r2s;
            const unsigned m = __builtin_amdgcn_ballot_w32(p);
            if (m) {
                const int lower = __popc(m & ltmask);
                if (c2 == 0) {
                    const int sl = __ffs((int)m) - 1;
                    f2x = __shfl(dx, sl, 32); f2y = __shfl(dy, sl, 32); f2z = __shfl(dz, sl, 32);
                }
                const int slot = c2 + lower;
                if (p && slot < 128) {
                    float* d = h2 + ((size_t)s * 128 + slot) * 3;
                    d[0] = dx; d[1] = dy; d[2] = dz;
                }
                c2 += __popc(m); if (c2 > 128) c2 = 128;
            }
        }
        { // radius 0.2 -> 32 samples
            const bool p = d2 <= r1s;
            const unsigned m = __builtin_amdgcn_ballot_w32(p);
            if (m) {
                const int lower = __popc(m & ltmask);
                if (c1 == 0) {
                    const int sl = __ffs((int)m) - 1;
                    f1x = __shfl(dx, sl, 32); f1y = __shfl(dy, sl, 32); f1z = __shfl(dz, sl, 32);
                }
                const int slot = c1 + lower;
                if (p && slot < 32) {
                    float* d = h1 + ((size_t)s * 32 + slot) * 3;
                    d[0] = dx; d[1] = dy; d[2] = dz;
                }
                c1 += __popc(m); if (c1 > 32) c1 = 32;
            }
        }
        { // radius 0.1 -> 16 samples
            const bool p = d2 <= r0s;
            const unsigned m = __builtin_amdgcn_ballot_w32(p);
            if (m) {
                const int lower = __popc(m & ltmask);
                if (c0 == 0) {
                    const int sl = __ffs((int)m) - 1;
                    f0x = __shfl(dx, sl, 32); f0y = __shfl(dy, sl, 32); f0z = __shfl(dz, sl, 32);
                }
                const int slot = c0 + lower;
                if (p && slot < 16) {
                    float* d = h0 + ((size_t)s * 16 + slot) * 3;
                    d[0] = dx; d[1] = dy; d[2] = dz;
                }
                c0 += __popc(m); if (c0 > 16) c0 = 16;
            }
        }
    }
    // pad unfilled slots with the first hit (center itself is always a hit)
    for (int j = c2 + lane; j < 128; j += 32) {
        float* d = h2 + ((size_t)s * 128 + j) * 3; d[0] = f2x; d[1] = f2y; d[2] = f2z;
    }
    for (int j = c1 + lane; j < 32; j += 32) {
        float* d = h1 + ((size_t)s * 32 + j) * 3; d[0] = f1x; d[1] = f1y; d[2] = f1z;
    }
    for (int j = c0 + lane; j < 16; j += 32) {
        float* d = h0 + ((size_t)s * 16 + j) * 3; d[0] = f0x; d[1] = f0y; d[2] = f0z;
    }
}

// ---------------------------------------------------------------------------
// WMMA f32 16x16x4 MLP layer: out = relu(in @ w + bias), LDS -> LDS.
// 128 rows per block, 4 waves, each wave owns M-tiles {wave, wave+4}.
// A layout: reg r -> K = r + 2*(lane>=16), M = lane%16.
// B/C/D layout: reg r -> N = lane%16, row = r (+2 for B / +8 for C/D on upper half).
// ---------------------------------------------------------------------------
__device__ __forceinline__ void mlp_layer(
    const float* __restrict__ w, const float* __restrict__ bias,
    int cin, int cout,
    const float* __restrict__ in, int istr,
    float* __restrict__ out, int ostr,
    float* __restrict__ wl, float* __restrict__ bl, int tid)
{
    const int Kpad = (cin + 3) & ~3;
    // stage zero-padded weights (Kpad x cout, stride 64) and bias in LDS
    for (int idx = tid; idx < Kpad * cout; idx += 128) {
        const int k = idx / cout, n = idx - k * cout;
        wl[k * 64 + n] = (k < cin) ? w[k * cout + n] : 0.0f;
    }
    if (tid < cout) bl[tid] = bias[tid];
    __syncthreads();

    const int lane = tid & 31;
    const int wv   = tid >> 5;
    const int l16  = lane & 15;
    const int hs2  = (lane >> 4) << 1;   // 0 (lanes 0-15) or 2 (lanes 16-31)

    for (int mtb = 0; mtb < 2; ++mtb) {
        const int mt  = wv + (mtb << 2);
        const int row = mt * 16 + l16;
        for (int nt = 0; nt < (cout >> 4); ++nt) {
            const float bv = bl[nt * 16 + l16];
            v8f c = { bv, bv, bv, bv, bv, bv, bv, bv };
            for (int kt = 0; kt < (Kpad >> 2); ++kt) {
                const int k0 = (kt << 2) + hs2;
                v2f a, bm;
                a.x  = in[row * istr + k0];
                a.y  = in[row * istr + k0 + 1];
                bm.x = wl[k0 * 64 + nt * 16 + l16];
                bm.y = wl[(k0 + 1) * 64 + nt * 16 + l16];
                c = __builtin_amdgcn_wmma_f32_16x16x4_f32(
                        false, a, false, bm, (short)0, c, false, false);
            }
#pragma unroll
            for (int r = 0; r < 8; ++r) {
                float vv = c[r];
                vv = vv > 0.0f ? vv : 0.0f;                       // fused ReLU
                out[(mt * 16 + hs2 * 4 + r) * ostr + nt * 16 + l16] = vv;
            }
        }
    }
    __syncthreads();
}

// ---------------------------------------------------------------------------
// Kernel 3: per-branch MLP + maxpool. 128 threads = 4 waves per block;
// each block handles CPB centers (CPB*NS == 128 rows).
// ---------------------------------------------------------------------------
template<int NS, int CPB, int C1, int C2, int C3>
__global__ __launch_bounds__(128) void branch_kernel(
    const float* __restrict__ h,
    const float* __restrict__ w1, const float* __restrict__ b1,
    const float* __restrict__ w2, const float* __restrict__ b2,
    const float* __restrict__ w3, const float* __restrict__ b3,
    float* __restrict__ feat)
{
    __shared__ float bufA[128 * 48];   // holds 3(pad4)-wide input, then C2<=48 output
    __shared__ float bufB[128 * 64];   // holds C1<=64 output, then C3<=64 output
    __shared__ float wl[48 * 64];
    __shared__ float bl[64];
    const int tid = threadIdx.x;
    const int cb  = blockIdx.x * CPB;

    { // load 128 rows of centered coords, K-pad col 3 with zero
        const float* src = h + ((size_t)cb * NS + tid) * 3;
        bufA[tid * 48 + 0] = src[0];
        bufA[tid * 48 + 1] = src[1];
        bufA[tid * 48 + 2] = src[2];
        bufA[tid * 48 + 3] = 0.0f;
    }
    __syncthreads();

    mlp_layer(w1, b1, 3,  C1, bufA, 48, bufB, 64, wl, bl, tid);
    mlp_layer(w2, b2, C1, C2, bufB, 64, bufA, 48, wl, bl, tid);
    mlp_layer(w3, b3, C2, C3, bufA, 48, bufB, 64, wl, bl, tid);

    // maxpool over NS samples per center (outputs are >=0 after ReLU)
    for (int idx = tid; idx < CPB * C3; idx += 128) {
        const int c = idx / C3, ch = idx - c * C3;
        float m = -1e30f;
        for (int r = 0; r < NS; ++r) m = fmaxf(m, bufB[(c * NS + r) * 64 + ch]);
        feat[((size_t)(cb + c)) * C3 + ch] = m;
    }
}

// ---------------------------------------------------------------------------
// Kernel 4: concat(32+64+64) @ w_out(160x1) + b_out
// ---------------------------------------------------------------------------
__global__ __launch_bounds__(256) void final_kernel(
    const float* __restrict__ f0, const float* __restrict__ f1,
    const float* __restrict__ f2, const float* __restrict__ wo,
    const float* __restrict__ bo, float* __restrict__ out)
{
    const int s = blockIdx.x * blockDim.x + threadIdx.x;
    if (s >= NCTOT) return;
    float acc = bo[0];
#pragma unroll 4
    for (int k = 0; k < 32; ++k) acc += f0[(size_t)s * 32 + k] * wo[k];
#pragma unroll 4
    for (int k = 0; k < 64; ++k) acc += f1[(size_t)s * 64 + k] * wo[32 + k];
#pragma unroll 4
    for (int k = 0; k < 64; ++k) acc += f2[(size_t)s * 64 + k] * wo[96 + k];
    out[s] = acc;
}

// ---------------------------------------------------------------------------
extern "C" void kernel_launch(void* const* d_in, const int* in_sizes, int n_in,
                              void* d_out, int out_size, void* d_ws, size_t ws_size,
                              hipStream_t stream) {
    const float* pcd = (const float*)d_in[0];
    // branch 0: d_in[1..6], branch 1: d_in[7..12], branch 2: d_in[13..18]
    const float* w0_0 = (const float*)d_in[1];  const float* b0_0 = (const float*)d_in[2];
    const float* w0_1 = (const float*)d_in[3];  const float* b0_1 = (const float*)d_in[4];
    const float* w0_2 = (const float*)d_in[5];  const float* b0_2 = (const float*)d_in[6];
    const float* w1_0 = (const float*)d_in[7];  const float* b1_0 = (const float*)d_in[8];
    const float* w1_1 = (const float*)d_in[9];  const float* b1_1 = (const float*)d_in[10];
    const float* w1_2 = (const float*)d_in[11]; const float* b1_2 = (const float*)d_in[12];
    const float* w2_0 = (const float*)d_in[13]; const float* b2_0 = (const float*)d_in[14];
    const float* w2_1 = (const float*)d_in[15]; const float* b2_1 = (const float*)d_in[16];
    const float* w2_2 = (const float*)d_in[17]; const float* b2_2 = (const float*)d_in[18];
    const float* w_out = (const float*)d_in[19];
    const float* b_out = (const float*)d_in[20];
    // d_in[21] = divide_ratio (unused by reference)

    // workspace layout (floats): ~11.3 MB total
    float* ws   = (float*)d_ws;
    float* nxyz = ws;                          // 4096*3
    float* h0   = nxyz + (size_t)NCTOT * 3;    // 4096*16*3
    float* h1   = h0 + (size_t)NCTOT * 16 * 3; // 4096*32*3
    float* h2   = h1 + (size_t)NCTOT * 32 * 3; // 4096*128*3
    float* f0   = h2 + (size_t)NCTOT * 128 * 3;// 4096*32
    float* f1   = f0 + (size_t)NCTOT * 32;     // 4096*64
    float* f2   = f1 + (size_t)NCTOT * 64;     // 4096*64
    float* out  = (float*)d_out;               // 4096 floats (B,1024,1)

    fps_kernel<<<NBATCH, 1024, 0, stream>>>(pcd, nxyz);
    ball_group_kernel<<<NCTOT / 8, 256, 0, stream>>>(pcd, nxyz, h0, h1, h2);
    branch_kernel<16, 8, 16, 16, 32><<<NCTOT / 8, 128, 0, stream>>>(
        h0, w0_0, b0_0, w0_1, b0_1, w0_2, b0_2, f0);
    branch_kernel<32, 4, 32, 32, 64><<<NCTOT / 4, 128, 0, stream>>>(
        h1, w1_0, b1_0, w1_1, b1_1, w1_2, b1_2, f1);
    branch_kernel<128, 1, 32, 48, 64><<<NCTOT, 128, 0, stream>>>(
        h2, w2_0, b2_0, w2_1, b2_1, w2_2, b2_2, f2);
    final_kernel<<<NCTOT / 256, 256, 0, stream>>>(f0, f1, f2, w_out, b_out, out);
}